// C_module_69131793596523
// MI455X (gfx1250) — compile-verified
//
#include <hip/hip_runtime.h>
#include <hip/hip_bf16.h>

// Shapes: x1 (16,32,256,256) f32 ; x2 (16,32,512,512) f32 ; out (16,32,256,256) f32
// BC = 512 planes.  mean1: BC*256 floats, mean2: BC*256 floats (both in d_ws).

typedef __attribute__((ext_vector_type(2))) float v2f;
typedef __attribute__((ext_vector_type(8))) float v8f;

#define BC   512
#define H1   256
#define W1   256
#define H2   512
#define W2   512

// ---------------------------------------------------------------------------
// Kernel 1: row means of x1.  One wave32 per row of 256 floats.
// Each lane does two b128 loads (8 floats), then a 5-step shuffle reduction.
// ---------------------------------------------------------------------------
__global__ void rowmean_x1_kernel(const float* __restrict__ x1,
                                  float* __restrict__ mean1) {
    const int wavesPerBlock = blockDim.x >> 5;
    const int row  = blockIdx.x * wavesPerBlock + (threadIdx.x >> 5); // 0..131071
    const int lane = threadIdx.x & 31;

    const float4* p = (const float4*)(x1 + (size_t)row * W1);
    float4 v0 = p[lane];        // cols [4*lane .. 4*lane+3]
    float4 v1 = p[lane + 32];   // cols [128+4*lane .. ]
    float s = (v0.x + v0.y + v0.z + v0.w) + (v1.x + v1.y + v1.z + v1.w);

    #pragma unroll
    for (int off = 16; off >= 1; off >>= 1)
        s += __shfl_xor(s, off, 32);

    if (lane == 0)
        mean1[row] = s * (1.0f / 256.0f);
}

// ---------------------------------------------------------------------------
// Kernel 2: column means of avg-pooled x2, folded:
//   mean2[bc][w] = (1/1024) * sum_{h=0..511} (x2[bc][h][2w] + x2[bc][h][2w+1])
// Thread q of plane bc streams columns [4q..4q+3] (one b128 per row, fully
// coalesced across the wave: 32 lanes * 16B = 512B contiguous) and produces
// two outputs (2q, 2q+1).  128 threads per plane.
// ---------------------------------------------------------------------------
__global__ void colmean_x2_kernel(const float* __restrict__ x2,
                                  float* __restrict__ mean2) {
    const int t  = blockIdx.x * blockDim.x + threadIdx.x; // 0..65535
    const int bc = t >> 7;
    const int q  = t & 127;

    const float4* p = (const float4*)(x2 + (size_t)bc * H2 * W2) + q;
    float a0 = 0.0f, a1 = 0.0f;
    #pragma unroll 4
    for (int h = 0; h < H2; ++h) {
        float4 v = p[(size_t)h * (W2 / 4)];
        a0 += v.x + v.y;   // columns 4q,4q+1  -> output 2q
        a1 += v.z + v.w;   // columns 4q+2,4q+3 -> output 2q+1
    }
    mean2[bc * W1 + 2 * q]     = a0 * (1.0f / 1024.0f);
    mean2[bc * W1 + 2 * q + 1] = a1 * (1.0f / 1024.0f);
}

// ---------------------------------------------------------------------------
// Kernel 3: rank-1 outer product epilogue via V_WMMA_F32_16X16X4_F32.
//   out_tile(16x16) = A(16x4) * B(4x16),  A col0 = mean1 seg, B row0 = mean2 seg.
// A layout (32-bit 16x4): lanes 0-15 hold M=0..15; VGPR0=K0, VGPR1=K1;
//                         lanes 16-31 hold K2,K3  -> we zero those.
// B layout (4x16): rows striped across lanes within a VGPR; row K=0 lives in
//                  VGPR0 lanes 0-15 -> mean2 there, everything else zero.
// D layout (16x16 f32): VGPR j = rows j (lanes 0-15) and j+8 (lanes 16-31).
// Block: 128 threads = 4 waves; each wave owns 4 of the 16 column tiles.
// Grid: (BC, H1/16).
// ---------------------------------------------------------------------------
__global__ void outer_wmma_kernel(const float* __restrict__ mean1,
                                  const float* __restrict__ mean2,
                                  float* __restrict__ out) {
    const int bc   = blockIdx.x;          // 0..511
    const int h0   = blockIdx.y * 16;     // 0..240
    const int wave = threadIdx.x >> 5;    // 0..3
    const int lane = threadIdx.x & 31;

    const float* m1 = mean1 + bc * H1 + h0;
    const float* m2 = mean2 + bc * W1;
    float*       o  = out + (size_t)bc * H1 * W1;

    const int lo16 = lane & 15;
    const bool lowHalf = (lane < 16);

    // A operand: column 0 = mean1[h0..h0+15], columns 1..3 = 0.
    float a_val = m1[lo16];               // in-range load for every lane
    v2f a;
    a.x = lowHalf ? a_val : 0.0f;         // K=0 (lanes 0-15) / K=2 (zeroed)
    a.y = 0.0f;                           // K=1 / K=3

    const int mbase = h0 + ((lane >> 4) << 3);  // rows j (+8 for upper half)

    #pragma unroll
    for (int i = 0; i < 4; ++i) {
        const int w0 = (wave * 4 + i) * 16;     // column tile origin

        // B operand: row 0 = mean2[w0..w0+15], rows 1..3 = 0.
        float b_val = m2[w0 + lo16];
        v2f b;
        b.x = lowHalf ? b_val : 0.0f;     // K=0 row (lanes 0-15); K=2 zeroed
        b.y = 0.0f;                       // K=1 / K=3 rows

        v8f c = {};
        v8f d = __builtin_amdgcn_wmma_f32_16x16x4_f32(
            /*neg_a=*/false, a, /*neg_b=*/false, b,
            /*c_mod=*/(short)0, c, /*reuse_a=*/false, /*reuse_b=*/false);

        const int n = w0 + lo16;
        #pragma unroll
        for (int j = 0; j < 8; ++j)
            o[(size_t)(mbase + j) * W1 + n] = d[j];
    }
}

// ---------------------------------------------------------------------------
extern "C" void kernel_launch(void* const* d_in, const int* in_sizes, int n_in,
                              void* d_out, int out_size, void* d_ws, size_t ws_size,
                              hipStream_t stream) {
    const float* x1 = (const float*)d_in[0];   // 16*32*256*256
    const float* x2 = (const float*)d_in[1];   // 16*32*512*512
    float* out = (float*)d_out;                // 16*32*256*256

    float* mean1 = (float*)d_ws;               // BC*H1 floats (512 KB)
    float* mean2 = mean1 + BC * H1;            // BC*W1 floats (512 KB)

    // 131072 rows, 8 waves per 256-thread block -> 16384 blocks
    rowmean_x1_kernel<<<(BC * H1) / 8, 256, 0, stream>>>(x1, mean1);

    // 512 planes * 128 threads = 65536 threads -> 256 blocks of 256
    colmean_x2_kernel<<<(BC * 128) / 256, 256, 0, stream>>>(x2, mean2);

    // Rank-1 WMMA epilogue: grid (512, 16), 128 threads (4 waves) per block
    dim3 grid(BC, H1 / 16);
    outer_wmma_kernel<<<grid, 128, 0, stream>>>(mean1, mean2, out);
}